// Model_79121887527001
// MI455X (gfx1250) — compile-verified
//
#include <hip/hip_runtime.h>
#include <hip/hip_bf16.h>
#include <math.h>

typedef float v2f __attribute__((ext_vector_type(2)));
typedef float v8f __attribute__((ext_vector_type(8)));

#define T_STEPS 8192
#define IN_DIM  512
#define H_DIM   1024
#define G4      4096          // 4*H
#define NWG     128           // persistent workgroups
#define UNITS   8             // hidden units per WG  (NWG*UNITS == H)
#define WROWS   32            // 4 gates * UNITS rows of W_hh per WG
#define LSTM_THREADS 256

// d_ws layout (in floats)
#define XPROJ_OFF 0                                   // T*4H floats (134 MB)
#define HBUF_OFF  ((size_t)T_STEPS * G4)              // 2*H floats (ping/pong h)
#define PRED_OFF  (HBUF_OFF + 2 * H_DIM)              // T*NWG partials (4 MB)
#define FLAG_OFF  (PRED_OFF + (size_t)T_STEPS * NWG)  // NWG uints (barrier flags)

// LDS byte offsets inside the dynamic segment (starts at 0: no static __shared__)
#define LDS_W_BYTES   (WROWS * H_DIM * 4)             // 131072
#define LDS_H_OFF     LDS_W_BYTES

// ---------------------------------------------------------------------------
// Init: zero h ping/pong and barrier flags each launch (graph-replay safe).
// ---------------------------------------------------------------------------
__global__ void init_ws(float* __restrict__ hbufs, unsigned int* __restrict__ flags) {
    int i = blockIdx.x * blockDim.x + threadIdx.x;
    if (i < 2 * H_DIM) hbufs[i] = 0.0f;
    if (i < NWG)       flags[i] = 0u;
}

// ---------------------------------------------------------------------------
// x_proj = seasons @ W_ih^T + b_ih + b_hh  via V_WMMA_F32_16X16X4_F32.
// Each wave computes a 16(M) x 64(N) strip: one A fragment reused for 4 B tiles.
// ---------------------------------------------------------------------------
__global__ __launch_bounds__(128)
void xproj_wmma(const float* __restrict__ S, const float* __restrict__ Wih,
                const float* __restrict__ bih, const float* __restrict__ bhh,
                float* __restrict__ xproj) {
    const int lane = threadIdx.x & 31;
    const int wave = threadIdx.x >> 5;
    const int m0 = (blockIdx.y * 4 + wave) * 16;
    const int n0 = blockIdx.x * 64;
    const int mrow = m0 + (lane & 15);
    const int koff = (lane >> 4) * 2;

    v8f acc[4] = {};
    const float* Srow = S + (size_t)mrow * IN_DIM + koff;

    for (int k0 = 0; k0 < IN_DIM; k0 += 4) {
        v2f a = *(const v2f*)(Srow + k0);
#pragma unroll
        for (int tt = 0; tt < 4; ++tt) {
            int ncol = n0 + tt * 16 + (lane & 15);
            v2f b = *(const v2f*)(Wih + (size_t)ncol * IN_DIM + k0 + koff);
            acc[tt] = __builtin_amdgcn_wmma_f32_16x16x4_f32(
                false, a, false, b, (short)0, acc[tt], false, false);
        }
    }

    const int mbase = m0 + (lane >> 4) * 8;
#pragma unroll
    for (int tt = 0; tt < 4; ++tt) {
        int ncol = n0 + tt * 16 + (lane & 15);
        float bias = bih[ncol] + bhh[ncol];
#pragma unroll
        for (int r = 0; r < 8; ++r)
            xproj[(size_t)(mbase + r) * G4 + ncol] = acc[tt][r] + bias;
    }
}

// ---------------------------------------------------------------------------
// Persistent LSTM recurrence. NWG WGs x 256 threads; each WG owns UNITS hidden
// units; its 32 W_hh rows (128 KB) stay LDS-resident for all 8192 steps.
// W_hh slice and per-step h broadcast staged with async LDS loads (ASYNCcnt).
// Grid sync per step: per-WG monotonic flags (no same-address atomic storm).
// ---------------------------------------------------------------------------
__device__ __forceinline__ float sigmoidf_(float x) {
    return 1.0f / (1.0f + expf(-x));
}

__global__ __launch_bounds__(LSTM_THREADS)
void lstm_persistent(const float* __restrict__ xproj, const float* __restrict__ Whh,
                     const float* __restrict__ Wlin,
                     float* __restrict__ hbufs, float* __restrict__ pred_part,
                     unsigned int* __restrict__ flags) {
    extern __shared__ float smem[];
    float* Wlds = smem;                       // [WROWS][1024], row r = gate*UNITS + u
    float* hlds = smem + WROWS * H_DIM;       // [1024]
    float* zbuf = hlds + H_DIM;               // [WROWS]
    float* cbuf = zbuf + WROWS;               // [UNITS]
    float* hwl  = cbuf + UNITS;               // [UNITS] pred partials

    const int g   = blockIdx.x;
    const int tid = threadIdx.x;

    // ---- Stage this WG's W_hh rows into LDS via async copy (memory -> LDS).
    // Per gate, the UNITS rows are contiguous in memory (32 KB each).
#pragma unroll
    for (int gate = 0; gate < 4; ++gate) {
        const float* gsrc = Whh + ((size_t)gate * H_DIM + g * UNITS) * H_DIM;
        const unsigned lbase = (unsigned)(gate * UNITS * H_DIM * 4);
        for (int i = tid; i < UNITS * H_DIM / 4; i += LSTM_THREADS) {
            unsigned loff = lbase + (unsigned)i * 16u;
            const float* src = gsrc + i * 4;
            asm volatile("global_load_async_to_lds_b128 %0, %1, off"
                         :: "v"(loff), "v"(src) : "memory");
        }
    }
    float wl = 0.0f;
    if (tid < UNITS) { cbuf[tid] = 0.0f; wl = Wlin[g * UNITS + tid]; }
    asm volatile("s_wait_asynccnt 0" ::: "memory");
    __syncthreads();

    const int lane = tid & 31;
    const int wave = tid >> 5;

    for (int t = 0; t < T_STEPS; ++t) {
        const float* hin  = hbufs + (t & 1) * H_DIM;         // h(t), zeros at t=0
        float*       hout = hbufs + ((t + 1) & 1) * H_DIM;   // h(t+1)

        // ---- Stage h(t) into LDS: 256 lanes x 16B async = 4 KB.
        {
            unsigned loff = (unsigned)LDS_H_OFF + (unsigned)tid * 16u;
            const float* src = hin + tid * 4;
            asm volatile("global_load_async_to_lds_b128 %0, %1, off"
                         :: "v"(loff), "v"(src) : "memory");
            asm volatile("s_wait_asynccnt 0" ::: "memory");
        }
        __syncthreads();

        // ---- 32 dot products of length 1024: 8 waves x 4 rows, float4 LDS reads.
#pragma unroll
        for (int rr = 0; rr < 4; ++rr) {
            int r = wave * 4 + rr;
            const float4* wrow4 = (const float4*)(Wlds + r * H_DIM);
            const float4* h4    = (const float4*)hlds;
            float s = 0.0f;
#pragma unroll
            for (int i = 0; i < 8; ++i) {
                int k = lane + i * 32;
                float4 w  = wrow4[k];
                float4 hh = h4[k];
                s += w.x * hh.x + w.y * hh.y + w.z * hh.z + w.w * hh.w;
            }
#pragma unroll
            for (int off = 16; off > 0; off >>= 1)
                s += __shfl_xor(s, off, 32);
            if (lane == 0) zbuf[r] = s;
        }
        __syncthreads();

        // ---- Gate math + state update for this WG's UNITS units.
        if (tid < UNITS) {
            int col = g * UNITS + tid;
            const float* xp = xproj + (size_t)t * G4;
            float zi = zbuf[tid]             + xp[col];
            float zf = zbuf[UNITS + tid]     + xp[H_DIM + col];
            float zg = zbuf[2 * UNITS + tid] + xp[2 * H_DIM + col];
            float zo = zbuf[3 * UNITS + tid] + xp[3 * H_DIM + col];
            float cn = sigmoidf_(zf) * cbuf[tid] + sigmoidf_(zi) * tanhf(zg);
            float hn = sigmoidf_(zo) * tanhf(cn);
            cbuf[tid] = cn;
            hout[col] = hn;          // publish h(t+1) slice
            hwl[tid]  = hn * wl;     // W_lin partial, reduced below
        }
        __threadfence();             // h(t+1) stores device-visible before arrive
        __syncthreads();

        if (tid == 0) {
            // One plain store per WG per step; summed by finalize kernel.
            float p = 0.0f;
#pragma unroll
            for (int u = 0; u < UNITS; ++u) p += hwl[u];
            pred_part[(size_t)t * NWG + g] = p;
            // Arrive: monotonic per-WG flag, release semantics.
            __hip_atomic_store(&flags[g], (unsigned)(t + 1),
                               __ATOMIC_RELEASE, __HIP_MEMORY_SCOPE_AGENT);
        }
        // Wait: thread i polls flag i (parallel, no RMW serialization).
        if (tid < NWG) {
            while (__hip_atomic_load(&flags[tid], __ATOMIC_ACQUIRE,
                                     __HIP_MEMORY_SCOPE_AGENT) < (unsigned)(t + 1))
                __builtin_amdgcn_s_sleep(1);
        }
        __syncthreads();
    }
}

// ---------------------------------------------------------------------------
// Finalize: preds = relu(sum(partials) + b_lin), apply mask, emit outputs.
// ---------------------------------------------------------------------------
__global__ void finalize_k(const float* __restrict__ seasons, const float* __restrict__ targets,
                           const float* __restrict__ pred_part, const float* __restrict__ b_lin,
                           float* __restrict__ out) {
    int t = blockIdx.x * blockDim.x + threadIdx.x;
    if (t >= T_STEPS) return;
    bool mask = true;
    const float* srow = seasons + (size_t)t * IN_DIM;
    for (int k = 0; k < IN_DIM; ++k) mask &= (srow[k] > -1.0f);
    float p = 0.0f;
    const float* pp = pred_part + (size_t)t * NWG;
    for (int i = 0; i < NWG; ++i) p += pp[i];
    p += b_lin[0];
    p = p > 0.0f ? p : 0.0f;
    out[t]           = mask ? p : 0.0f;          // preds [T,1]
    out[T_STEPS + t] = mask ? targets[t] : 0.0f; // targets_m [T]
}

// ---------------------------------------------------------------------------
extern "C" void kernel_launch(void* const* d_in, const int* in_sizes, int n_in,
                              void* d_out, int out_size, void* d_ws, size_t ws_size,
                              hipStream_t stream) {
    const float* seasons = (const float*)d_in[0];
    const float* targets = (const float*)d_in[1];
    const float* W_ih    = (const float*)d_in[2];
    const float* W_hh    = (const float*)d_in[3];
    const float* b_ih    = (const float*)d_in[4];
    const float* b_hh    = (const float*)d_in[5];
    const float* W_lin   = (const float*)d_in[6];
    const float* b_lin   = (const float*)d_in[7];
    float* out = (float*)d_out;

    float* ws         = (float*)d_ws;
    float* xproj      = ws + XPROJ_OFF;
    float* hbufs      = ws + HBUF_OFF;
    float* pred_part  = ws + PRED_OFF;
    unsigned int* flags = (unsigned int*)(ws + FLAG_OFF);

    init_ws<<<(2 * H_DIM + 255) / 256, 256, 0, stream>>>(hbufs, flags);

    dim3 ggrid(G4 / 64, T_STEPS / 16 / 4);
    xproj_wmma<<<ggrid, 128, 0, stream>>>(seasons, W_ih, b_ih, b_hh, xproj);

    size_t lds_bytes = (size_t)(WROWS * H_DIM + H_DIM + WROWS + UNITS + UNITS) * sizeof(float);
    lstm_persistent<<<NWG, LSTM_THREADS, lds_bytes, stream>>>(
        xproj, W_hh, W_lin, hbufs, pred_part, flags);

    finalize_k<<<(T_STEPS + 255) / 256, 256, 0, stream>>>(
        seasons, targets, pred_part, b_lin, out);
}